// MS_27582279975040
// MI455X (gfx1250) — compile-verified
//
#include <hip/hip_runtime.h>

// Contrastive embedding loss, fused: never materialize the 4096x4096 sim matrix.
// sim tile (16x16) computed with chained V_WMMA_F32_16X16X4_F32 (K=16 -> 4 WMMAs),
// loss epilogue folded onto the accumulator fragment, symmetric-half tiling.

typedef float v2f __attribute__((ext_vector_type(2)));
typedef float v8f __attribute__((ext_vector_type(8)));

#define NPIX  4096      // 64*64 pixels per batch image
#define NCH   16        // channels
#define NB    4         // batch
#define NLAB  20        // label ids
#define TILES 256       // NPIX / 16

// workspace layout (in 4-byte units)
#define OFF_E   0                       // normalized embeddings, row-major [b][pix][ch]
#define OFF_CNT (NB * NPIX * NCH)       // per-batch label histogram [b][20]
#define OFF_LW  (OFF_CNT + NB * NLAB)   // packed per-pixel {label, bitcast(weight)} uint2

// Single v_max_num_f32 with inline-constant 0. Forced via asm so LLVM cannot
// re-insert the canonicalize (v_max v,v,v) it pairs with fmaxf/fmed3 lowering.
// WMMA outputs are canonical (RNE, no sNaN), so this is exact relu.
__device__ __forceinline__ float relu_raw(float x) {
    float r;
    asm("v_max_num_f32 %0, %1, 0" : "=v"(r) : "v"(x));
    return r;
}

__global__ void k_zero(float* __restrict__ out, float* __restrict__ cnt) {
    int t = threadIdx.x;
    if (t < NB * NLAB) cnt[t] = 0.0f;
    if (t == NB * NLAB) out[0] = 0.0f;
}

// Per-pixel: center across channels, L2-normalize (NaN-masked), store row-major;
// also build per-batch label histogram.
__global__ void k_prep(const float* __restrict__ emb, const int* __restrict__ lab,
                       float* __restrict__ e, float* __restrict__ cnt) {
    int p = blockIdx.x * blockDim.x + threadIdx.x;   // [0, NB*NPIX)
    if (p >= NB * NPIX) return;
    int b = p >> 12;
    int n = p & (NPIX - 1);
    const float* x = emb + (size_t)b * NCH * NPIX + n;   // channel-strided

    float v[NCH];
    float mean = 0.0f;
    #pragma unroll
    for (int c = 0; c < NCH; ++c) { v[c] = x[(size_t)c * NPIX]; mean += v[c]; }
    mean *= (1.0f / NCH);
    float s2 = 0.0f;
    #pragma unroll
    for (int c = 0; c < NCH; ++c) { v[c] -= mean; s2 += v[c] * v[c]; }
    float nrm = sqrtf(s2);
    float inv = (nrm > 0.0f) ? (1.0f / nrm) : 0.0f;   // reference NaN-mask -> 0

    float* er = e + (size_t)p * NCH;
    #pragma unroll
    for (int c = 0; c < NCH; ++c) er[c] = v[c] * inv;

    atomicAdd(&cnt[b * NLAB + lab[p]], 1.0f);
}

// Pack per-pixel {label, inverse-frequency weight} into one 8-byte record so the
// main loop fetches both with a single global_load_b64.
__global__ void k_lw(const int* __restrict__ lab, const float* __restrict__ cnt,
                     uint2* __restrict__ lw) {
    int p = blockIdx.x * blockDim.x + threadIdx.x;
    if (p >= NB * NPIX) return;
    int b = p >> 12;
    float w = 1.0f / cnt[b * NLAB + lab[p]];   // only present labels looked up -> cnt >= 1
    lw[p] = make_uint2((unsigned)lab[p], __float_as_uint(w));
}

// One block per (tile-row ti, batch b); 8 waves stride over tile-cols tj >= ti.
// f32 WMMA fragment formula (per CDNA5 ISA layouts):
//   A 16x4:  lane(l): m=l&15, h=l>>4 -> VGPR0/1 = e[i+m][4k+2h], e[i+m][4k+2h+1]
//   B 4x16:  identical with i->j (since B = e-tile transposed)
//   C/D f32: VGPR v on lane l -> element (M = v + 8h, N = l&15)
__global__ void __launch_bounds__(256)
k_loss(const float* __restrict__ e, const uint2* __restrict__ lw,
       float* __restrict__ out) {
    const int ti = blockIdx.x;
    const int b  = blockIdx.y;
    const int wv = threadIdx.x >> 5;     // wave id 0..7 (wave32)
    const int l  = threadIdx.x & 31;
    const int m  = l & 15;
    const int h  = l >> 4;

    const float* eb  = e  + (size_t)b * NPIX * NCH;
    const uint2* lwb = lw + (size_t)b * NPIX;

    const int i = ti * 16;

    // A fragments fixed for the whole block (row tile is fixed): load once.
    const v2f a0 = *(const v2f*)&eb[(i + m) * NCH +  0 + 2 * h];
    const v2f a1 = *(const v2f*)&eb[(i + m) * NCH +  4 + 2 * h];
    const v2f a2 = *(const v2f*)&eb[(i + m) * NCH +  8 + 2 * h];
    const v2f a3 = *(const v2f*)&eb[(i + m) * NCH + 12 + 2 * h];

    // Row-side labels/weights for the 8 D-matrix rows this lane owns.
    unsigned rl[8];
    float    rw[8];
    #pragma unroll
    for (int v = 0; v < 8; ++v) {
        uint2 q = lwb[i + v + 8 * h];
        rl[v] = q.x;
        rw[v] = __uint_as_float(q.y);
    }

    float acc = 0.0f;
    for (int tj = ti + wv; tj < TILES; tj += 8) {   // uniform per wave: EXEC all-ones
        const int j = tj * 16;
        v2f b0 = *(const v2f*)&eb[(j + m) * NCH +  0 + 2 * h];
        v2f b1 = *(const v2f*)&eb[(j + m) * NCH +  4 + 2 * h];
        v2f b2 = *(const v2f*)&eb[(j + m) * NCH +  8 + 2 * h];
        v2f b3 = *(const v2f*)&eb[(j + m) * NCH + 12 + 2 * h];

        v8f cc = {};
        cc = __builtin_amdgcn_wmma_f32_16x16x4_f32(false, a0, false, b0, (short)0, cc, false, false);
        cc = __builtin_amdgcn_wmma_f32_16x16x4_f32(false, a1, false, b1, (short)0, cc, false, false);
        cc = __builtin_amdgcn_wmma_f32_16x16x4_f32(false, a2, false, b2, (short)0, cc, false, false);
        cc = __builtin_amdgcn_wmma_f32_16x16x4_f32(false, a3, false, b3, (short)0, cc, false, false);

        const uint2 q  = lwb[j + m];
        const unsigned cl = q.x;
        const float    cw = __uint_as_float(q.y);

        // loss(pair) = 0.5 * w_i*w_j * max(eq ? 1-dot : dot, 0)  (MARGIN = 0.5)
        float ts = 0.0f;
        #pragma unroll
        for (int v = 0; v < 8; ++v) {
            float cv = cc[v];
            float t  = (rl[v] == cl) ? (1.0f - cv) : cv;
            t = relu_raw(t);
            ts = fmaf(rw[v], t, ts);
        }
        // diag tile counted once (x0.5 overall), off-diag twice (0.5*2 = 1)
        acc = fmaf((tj == ti) ? 0.5f * cw : cw, ts, acc);
    }

    // wave32 reduction, then one atomic per wave
    #pragma unroll
    for (int o = 16; o > 0; o >>= 1) acc += __shfl_xor(acc, o, 32);
    if (l == 0) atomicAdd(out, acc * (1.0f / NPIX));
}

extern "C" void kernel_launch(void* const* d_in, const int* in_sizes, int n_in,
                              void* d_out, int out_size, void* d_ws, size_t ws_size,
                              hipStream_t stream) {
    (void)in_sizes; (void)n_in; (void)out_size; (void)ws_size;
    const float* emb = (const float*)d_in[0];   // (4, 16, 64, 64) fp32
    const int*   lab = (const int*)d_in[1];     // (4, 64, 64) int32
    float* out = (float*)d_out;                 // scalar
    float* ws  = (float*)d_ws;
    float* e   = ws + OFF_E;
    float* cnt = ws + OFF_CNT;
    uint2* lw  = (uint2*)(ws + OFF_LW);

    k_zero<<<1, 128, 0, stream>>>(out, cnt);
    k_prep<<<(NB * NPIX + 255) / 256, 256, 0, stream>>>(emb, lab, e, cnt);
    k_lw<<<(NB * NPIX + 255) / 256, 256, 0, stream>>>(lab, cnt, lw);
    dim3 grid(TILES, NB);
    k_loss<<<grid, 256, 0, stream>>>(e, lw, out);
}